// CustomLSTM_49692771615279
// MI455X (gfx1250) — compile-verified
//
#include <hip/hip_runtime.h>
#include <hip/hip_bf16.h>

typedef __attribute__((ext_vector_type(2))) float v2f;
typedef __attribute__((ext_vector_type(8))) float v8f;

#define LSTM_I 5
#define LSTM_H 5
#define LSTM_G 20   /* 4*H */
#define KDIM   10   /* I + H, padded to 12 across 3 K-tiles of 4 */
#define WARPS_PER_BLOCK 8
#define BATCH_PER_WARP 16
#define BATCH_PER_BLOCK (WARPS_PER_BLOCK * BATCH_PER_WARP)

__device__ __forceinline__ float tanh_fast(float x) {
#if __has_builtin(__builtin_amdgcn_tanhf)
  return __builtin_amdgcn_tanhf(x);            // v_tanh_f32 (gfx1250 TRANS op)
#else
  float e = __builtin_amdgcn_exp2f(-2.8853900817779268f * x);
  return 2.0f * __builtin_amdgcn_rcpf(1.0f + e) - 1.0f;
#endif
}
__device__ __forceinline__ float sig_fast(float x) {
  // sigmoid(x) = 0.5*tanh(0.5x) + 0.5  -> single TRANS op
  return fmaf(0.5f, tanh_fast(0.5f * x), 0.5f);
}

__global__ __launch_bounds__(256) void lstm_wmma_kernel(
    const float* __restrict__ x, const float* __restrict__ h0,
    const float* __restrict__ c0, const float* __restrict__ W_ih,
    const float* __restrict__ W_hh, const float* __restrict__ b_ih,
    const float* __restrict__ b_hh, const float* __restrict__ fc1_w,
    const float* __restrict__ fc1_b, const float* __restrict__ fc2_w,
    const float* __restrict__ fc2_b, float* __restrict__ out,
    int Bn, int Tn) {
  const int lane = threadIdx.x & 31;
  const int warp = threadIdx.x >> 5;
  const int col  = lane & 15;   // batch column (N index)
  const int half = lane >> 4;   // half-wave: selects K / M sub-rows

  int n = blockIdx.x * BATCH_PER_BLOCK + warp * BATCH_PER_WARP + col;
  if (n > Bn - 1) n = Bn - 1;   // clamp; keeps EXEC all-ones for WMMA

  // ---- A-tiles: Wcat = [W_ih | W_hh] (20 x 10), zero-padded to 32 x 12.
  v2f A[2][3];
#pragma unroll
  for (int mt = 0; mt < 2; ++mt) {
    int m = mt * 16 + col;
#pragma unroll
    for (int kt = 0; kt < 3; ++kt) {
      int k0 = kt * 4 + half * 2;
      float a0 = 0.f, a1 = 0.f;
      if (m < LSTM_G) {
        if (k0 < LSTM_I)      a0 = W_ih[m * LSTM_I + k0];
        else if (k0 < KDIM)   a0 = W_hh[m * LSTM_H + (k0 - LSTM_I)];
        int k1 = k0 + 1;
        if (k1 < LSTM_I)      a1 = W_ih[m * LSTM_I + k1];
        else if (k1 < KDIM)   a1 = W_hh[m * LSTM_H + (k1 - LSTM_I)];
      }
      A[mt][kt].x = a0;
      A[mt][kt].y = a1;
    }
  }

  // ---- bias in C-tile layout (row M = gate index, lane = batch col)
  v8f cb0, cb1;
#pragma unroll
  for (int r = 0; r < 8; ++r) {
    int g0 = half * 8 + r;
    cb0[r] = b_ih[g0] + b_hh[g0];
    int g1 = 16 + half * 8 + r;
    cb1[r] = (g1 < LSTM_G) ? (b_ih[g1] + b_hh[g1]) : 0.f;
  }

  // ---- split state: low half owns h/c {0,3,4}; high half owns {1,2} (+dummy)
  const int jA = half ? 1 : 0;
  const int jB = half ? 2 : 3;
  float hs0 = h0[(size_t)n * LSTM_H + jA];
  float hs1 = h0[(size_t)n * LSTM_H + jB];
  float hs2 = h0[(size_t)n * LSTM_H + 4];   // high-half value is don't-care
  float cs0 = c0[(size_t)n * LSTM_H + jA];
  float cs1 = c0[(size_t)n * LSTM_H + jB];
  float cs2 = c0[(size_t)n * LSTM_H + 4];

  // per-lane x pointers: low half loads x0,x1,x4; high half loads x2,x3,(x2 dummy)
  const float* xrow = x + (size_t)n * Tn * LSTM_I;
  const float* xq = xrow + (half ? 2 : 0);
  const float* xr = xrow + (half ? 2 : 4);
  float nxA = xq[0], nxB = xq[1], nxC = xr[0];

#pragma unroll 1
  for (int t = 0; t < Tn; ++t) {
    // ---- B-tiles (lane = batch col; K rows split by half-wave)
    // S = [x0..x4, h0..h4, 0, 0]
    v2f B0, B1, B2;
    B0.x = nxA;                    // K0: x0 | K2: x2
    B0.y = nxB;                    // K1: x1 | K3: x3
    B1.x = half ? hs0 : nxC;       // K4: x4 | K6: h1
    B1.y = half ? hs1 : hs0;       // K5: h0 | K7: h2
    B2.x = half ? 0.f : hs1;       // K8: h3 | K10: 0
    B2.y = half ? 0.f : hs2;       // K9: h4 | K11: 0

    // ---- software-pipelined loads for t+1 (hidden under WMMA + TRANS chain)
    int step = (t + 1 < Tn) ? LSTM_I : 0;
    xq += step;  xr += step;
    nxA = xq[0]; nxB = xq[1]; nxC = xr[0];
    // locality 3 -> WGP-scope prefetch: pull into all cache levels
    __builtin_prefetch(xq + 32 * LSTM_I, 0, 3);

    // ---- gates^T = Wcat * S (+bias): 2 M-tiles x 3 K-tiles
    v8f C0 = __builtin_amdgcn_wmma_f32_16x16x4_f32(false, A[0][0], false, B0, (short)0, cb0, false, false);
    C0     = __builtin_amdgcn_wmma_f32_16x16x4_f32(false, A[0][1], false, B1, (short)0, C0,  false, false);
    C0     = __builtin_amdgcn_wmma_f32_16x16x4_f32(false, A[0][2], false, B2, (short)0, C0,  false, false);
    v8f C1 = __builtin_amdgcn_wmma_f32_16x16x4_f32(false, A[1][0], false, B0, (short)0, cb1, false, false);
    C1     = __builtin_amdgcn_wmma_f32_16x16x4_f32(false, A[1][1], false, B1, (short)0, C1,  false, false);
    C1     = __builtin_amdgcn_wmma_f32_16x16x4_f32(false, A[1][2], false, B2, (short)0, C1,  false, false);

    // ---- exchange only the C registers the other half needs
    float o00 = __shfl_xor(C0[0], 16, 32);
    float o01 = __shfl_xor(C0[1], 16, 32);
    float o02 = __shfl_xor(C0[2], 16, 32);
    float o05 = __shfl_xor(C0[5], 16, 32);
    float o06 = __shfl_xor(C0[6], 16, 32);
    float o07 = __shfl_xor(C0[7], 16, 32);
    float o10 = __shfl_xor(C1[0], 16, 32);
    float o11 = __shfl_xor(C1[1], 16, 32);

    // ---- per-slot gate operands; low half: j={0,3,4}, high half: j={1,2,(dc)}
    float i0  = half ? o01   : C0[0];   // g[j]    : g1  | g0
    float f0  = half ? o06   : C0[5];   // g[5+j]  : g6  | g5
    float g0v = half ? C0[3] : o02;     // g[10+j] : g11 | g10
    float oo0 = half ? o10   : o07;     // g[15+j] : g16 | g15

    float i1  = half ? o02   : C0[3];   // g2  | g3
    float f1  = half ? o07   : o00;     // g7  | g8
    float g1v = half ? C0[4] : o05;     // g12 | g13
    float oo1 = half ? o11   : C1[2];   // g17 | g18

    float i2  = C0[4];                  // g4 (high half: don't-care)
    float f2  = o01;                    // g9
    float g2v = o06;                    // g14
    float oo2 = C1[3];                  // g19

    // ---- pointwise update (each instruction does different work per half)
    cs0 = sig_fast(f0) * cs0 + sig_fast(i0) * tanh_fast(g0v);
    hs0 = sig_fast(oo0) * tanh_fast(cs0);
    cs1 = sig_fast(f1) * cs1 + sig_fast(i1) * tanh_fast(g1v);
    hs1 = sig_fast(oo1) * tanh_fast(cs1);
    cs2 = sig_fast(f2) * cs2 + sig_fast(i2) * tanh_fast(g2v);
    hs2 = sig_fast(oo2) * tanh_fast(cs2);
  }

  // ---- FC heads: low half has h0,h3,h4; fetch h1,h2 from high half
  float ph1 = __shfl_xor(hs0, 16, 32);
  float ph2 = __shfl_xor(hs1, 16, 32);
  if (half == 0) {
    float p = fc1_b[0] + hs0 * fc1_w[0] + ph1 * fc1_w[1] + ph2 * fc1_w[2] +
              hs1 * fc1_w[3] + hs2 * fc1_w[4];
    float v = fc2_b[0] + hs0 * fc2_w[0] + ph1 * fc2_w[1] + ph2 * fc2_w[2] +
              hs1 * fc2_w[3] + hs2 * fc2_w[4];
    out[n]      = p;
    out[Bn + n] = v;
  }
}

extern "C" void kernel_launch(void* const* d_in, const int* in_sizes, int n_in,
                              void* d_out, int out_size, void* d_ws, size_t ws_size,
                              hipStream_t stream) {
  const float* x    = (const float*)d_in[0];
  const float* h0   = (const float*)d_in[1];
  const float* c0   = (const float*)d_in[2];
  const float* W_ih = (const float*)d_in[3];
  const float* W_hh = (const float*)d_in[4];
  const float* b_ih = (const float*)d_in[5];
  const float* b_hh = (const float*)d_in[6];
  const float* fc1w = (const float*)d_in[7];
  const float* fc1b = (const float*)d_in[8];
  const float* fc2w = (const float*)d_in[9];
  const float* fc2b = (const float*)d_in[10];
  float* out = (float*)d_out;

  int Bn = in_sizes[1] / LSTM_H;           // h0 = [B,H]
  int Tn = in_sizes[0] / (Bn * LSTM_I);    // x  = [B,T,I]

  int grid = (Bn + BATCH_PER_BLOCK - 1) / BATCH_PER_BLOCK;
  lstm_wmma_kernel<<<grid, 32 * WARPS_PER_BLOCK, 0, stream>>>(
      x, h0, c0, W_ih, W_hh, b_ih, b_hh, fc1w, fc1b, fc2w, fc2b, out, Bn, Tn);
}